// ModelTorch_87333864997435
// MI455X (gfx1250) — compile-verified
//
#include <hip/hip_runtime.h>
#include <hip/hip_bf16.h>

// ---------------------------------------------------------------------------
// Packed ragged gather (speculative-decode cache-loc packing) for gfx1250.
//
//   tgt_cache_loc  = concat_b( out_cache_loc[b, 0 : L[b]] ),   L = accept+1
//   to_free_slots  = concat_b( out_cache_loc[b, V-F[b] : V] ), F = to_free
//
// Pipeline:
//   K1 scan_rows:   per-256-row block scan of L and F -> in-block offsets + block sums
//   K2 scan_blocks: single-block exclusive scan of the 128 block sums + totals
//   K3 copy_rows:   1 workgroup per row; async global->LDS->global DMA copy
//                   (CDNA5 GLOBAL_LOAD_ASYNC_TO_LDS_B32 / STORE_ASYNC_FROM_LDS_B32,
//                    ASYNCcnt-tracked), coalesced b32 fallback otherwise.
// ---------------------------------------------------------------------------

#define RPB 256   // rows per scan block (kernel 1)
#define CPT 128   // copy threads per row block (kernel 3)

#define AS1 __attribute__((address_space(1)))
#define AS3 __attribute__((address_space(3)))

#if defined(__AMDGCN__) && defined(__has_builtin)
#  if __has_builtin(__builtin_amdgcn_global_load_async_to_lds_b32) && \
      __has_builtin(__builtin_amdgcn_global_store_async_from_lds_b32)
#    define USE_ASYNC_LDS 1
#  endif
#endif
#ifndef USE_ASYNC_LDS
#  define USE_ASYNC_LDS 0
#endif

__device__ __forceinline__ void wait_async0() {
#if defined(__AMDGCN__)
#  if defined(__has_builtin) && __has_builtin(__builtin_amdgcn_s_wait_asynccnt)
  __builtin_amdgcn_s_wait_asynccnt(0);
#  else
  asm volatile("s_wait_asynccnt 0" ::: "memory");
#  endif
#endif
}

// ---- Kernel 1: per-block inclusive scan of L=acc+1 and F over 256 rows ----
__global__ void scan_rows_kernel(const int* __restrict__ acc,
                                 const int* __restrict__ fre,
                                 unsigned* __restrict__ inOffL,
                                 unsigned* __restrict__ inOffF,
                                 unsigned* __restrict__ blkL,
                                 unsigned* __restrict__ blkF,
                                 int B) {
  __shared__ unsigned sL[RPB], sF[RPB];
  const int t = threadIdx.x;
  const int r = blockIdx.x * RPB + t;
  unsigned L = 0u, F = 0u;
  if (r < B) {
    L = (unsigned)(acc[r] + 1);
    F = (unsigned)fre[r];
  }
  sL[t] = L; sF[t] = F;
  __syncthreads();
  for (int off = 1; off < RPB; off <<= 1) {
    unsigned aL = (t >= off) ? sL[t - off] : 0u;
    unsigned aF = (t >= off) ? sF[t - off] : 0u;
    __syncthreads();
    sL[t] += aL; sF[t] += aF;
    __syncthreads();
  }
  if (r < B) {
    inOffL[r] = sL[t] - L;   // exclusive within block
    inOffF[r] = sF[t] - F;
  }
  if (t == RPB - 1) {
    blkL[blockIdx.x] = sL[t];
    blkF[blockIdx.x] = sF[t];
  }
}

// ---- Kernel 2: exclusive scan of block sums (single block, pow2 threads) ----
__global__ void scan_blocks_kernel(unsigned* __restrict__ blkL,
                                   unsigned* __restrict__ blkF,
                                   unsigned* __restrict__ totals,
                                   int NB) {
  __shared__ unsigned sL[1024], sF[1024];
  const int t = threadIdx.x;
  const int n = blockDim.x;
  unsigned L = (t < NB) ? blkL[t] : 0u;
  unsigned F = (t < NB) ? blkF[t] : 0u;
  sL[t] = L; sF[t] = F;
  __syncthreads();
  for (int off = 1; off < n; off <<= 1) {
    unsigned aL = (t >= off) ? sL[t - off] : 0u;
    unsigned aF = (t >= off) ? sF[t - off] : 0u;
    __syncthreads();
    sL[t] += aL; sF[t] += aF;
    __syncthreads();
  }
  if (t < NB) {
    blkL[t] = sL[t] - L;     // exclusive block offset
    blkF[t] = sF[t] - F;
  }
  if (t == n - 1) {
    totals[0] = sL[t];       // total_tgt -> start of to_free region in d_out
    totals[1] = sF[t];
  }
}

// ---- Kernel 3: one workgroup per row; async DMA copy through LDS ----------
__global__ void copy_rows_kernel(const int* __restrict__ acc,
                                 const int* __restrict__ fre,
                                 const int* __restrict__ flat,
                                 const unsigned* __restrict__ inOffL,
                                 const unsigned* __restrict__ inOffF,
                                 const unsigned* __restrict__ blkL,
                                 const unsigned* __restrict__ blkF,
                                 const unsigned* __restrict__ totals,
                                 int* __restrict__ out,
                                 int V, int chunks) {
  extern __shared__ int lds[];
  const int b  = blockIdx.x;
  const int t  = threadIdx.x;
  const int nt = blockDim.x;                 // CPT
  const int L  = acc[b] + 1;
  const int F  = fre[b];
  const unsigned tgt0 = blkL[b / RPB] + inOffL[b];
  const unsigned fre0 = totals[0] + blkF[b / RPB] + inOffF[b];
  const int* row     = flat + (long long)b * V;
  const int* rowTail = row + (V - F);

#if USE_ASYNC_LDS
  // Issue all loads (up to 2*chunks outstanding per wave), one wait, all stores.
  // Each thread owns private LDS slots, so per-wave ASYNCcnt ordering suffices
  // (no workgroup barrier needed). Stores drain at s_endpgm's implicit wait.
  for (int k = 0; k < chunks; ++k) {
    const int i = k * nt + t;
    if (i < L)
      __builtin_amdgcn_global_load_async_to_lds_b32(
          (AS1 int*)(row + i), (AS3 int*)&lds[k * nt + t], 0, 0);
    if (i < F)
      __builtin_amdgcn_global_load_async_to_lds_b32(
          (AS1 int*)(rowTail + i), (AS3 int*)&lds[(chunks + k) * nt + t], 0, 0);
  }
  wait_async0();
  for (int k = 0; k < chunks; ++k) {
    const int i = k * nt + t;
    if (i < L)
      __builtin_amdgcn_global_store_async_from_lds_b32(
          (AS1 int*)(out + tgt0 + i), (AS3 int*)&lds[k * nt + t], 0, 0);
    if (i < F)
      __builtin_amdgcn_global_store_async_from_lds_b32(
          (AS1 int*)(out + fre0 + i), (AS3 int*)&lds[(chunks + k) * nt + t], 0, 0);
  }
#else
  // Coalesced b32 fallback (still memory-optimal; whole set fits in 192MB L2).
  for (int i = t; i < L; i += nt) out[tgt0 + i] = row[i];
  for (int i = t; i < F; i += nt) out[fre0 + i] = rowTail[i];
#endif
}

// ---------------------------------------------------------------------------
extern "C" void kernel_launch(void* const* d_in, const int* in_sizes, int n_in,
                              void* d_out, int out_size, void* d_ws, size_t ws_size,
                              hipStream_t stream) {
  const int* acc  = (const int*)d_in[0];   // accept_length      [B]
  const int* fre  = (const int*)d_in[1];   // to_free_num_slots  [B]
  const int* flat = (const int*)d_in[2];   // out_cache_loc      [B*V]
  int*       out  = (int*)d_out;

  const int B  = in_sizes[0];
  const int V  = in_sizes[2] / in_sizes[0];
  const int NB = (B + RPB - 1) / RPB;

  // workspace layout (unsigned words)
  unsigned* ws     = (unsigned*)d_ws;
  unsigned* inOffL = ws;                   // [B]
  unsigned* inOffF = inOffL + B;           // [B]
  unsigned* blkL   = inOffF + B;           // [NB]
  unsigned* blkF   = blkL + NB;            // [NB]
  unsigned* totals = blkF + NB;            // [2]

  scan_rows_kernel<<<NB, RPB, 0, stream>>>(acc, fre, inOffL, inOffF, blkL, blkF, B);

  int p2 = 1;
  while (p2 < NB) p2 <<= 1;
  if (p2 > 1024) p2 = 1024;                // B <= 262144 assumed
  scan_blocks_kernel<<<1, p2, 0, stream>>>(blkL, blkF, totals, NB);

  const int    chunks = (V + CPT - 1) / CPT;
  const size_t smem   = (size_t)2 * chunks * CPT * sizeof(int);
  copy_rows_kernel<<<B, CPT, smem, stream>>>(acc, fre, flat, inOffL, inOffF,
                                             blkL, blkF, totals, out, V, chunks);
}